// GCN2_56307021250667
// MI455X (gfx1250) — compile-verified
//
#include <hip/hip_runtime.h>
#include <hip/hip_bf16.h>

#define N_NODES 50000
#define N_EDGES 800000
#define BN_EPS 1e-5f
#define KC 32  // K-chunk staged in LDS per iteration

typedef __attribute__((ext_vector_type(2))) float v2f;
typedef __attribute__((ext_vector_type(8))) float v8f;
typedef __attribute__((ext_vector_type(4))) int   v4i;

#if __has_builtin(__builtin_amdgcn_global_load_async_to_lds_b128) && \
    __has_builtin(__builtin_amdgcn_s_wait_asynccnt)
#define USE_ASYNC_LDS 1
typedef __attribute__((address_space(1))) v4i* gas_v4i_ptr;
typedef __attribute__((address_space(3))) v4i* las_v4i_ptr;
#else
#define USE_ASYNC_LDS 0
#endif

// ---------------- degree / norm precompute ----------------

__global__ void deg_init(float* __restrict__ dinv, int n) {
  int i = blockIdx.x * blockDim.x + threadIdx.x;
  if (i < n) dinv[i] = 1.0f;  // self-loop contributes 1 to degree
}

__global__ void deg_accum(const int* __restrict__ col, float* __restrict__ dinv, int e) {
  int i = blockIdx.x * blockDim.x + threadIdx.x;
  if (i < e) atomicAdd(&dinv[col[i]], 1.0f);
}

__global__ void deg_fin(float* __restrict__ dinv, int n) {
  int i = blockIdx.x * blockDim.x + threadIdx.x;
  if (i < n) dinv[i] = rsqrtf(dinv[i]);
}

__global__ void edge_norm(const int* __restrict__ row, const int* __restrict__ col,
                          const float* __restrict__ dinv, float* __restrict__ nrm, int e) {
  int i = blockIdx.x * blockDim.x + threadIdx.x;
  if (i < e) nrm[i] = dinv[row[i]] * dinv[col[i]];
}

// ---------------- BatchNorm (eval) over up to 3 concatenated sources ----------------

__global__ void bn_eval_concat(const float* __restrict__ s0, const float* __restrict__ s1,
                               const float* __restrict__ s2, int d0, int d1, int d2,
                               const float* __restrict__ g, const float* __restrict__ b,
                               const float* __restrict__ rm, const float* __restrict__ rv,
                               float* __restrict__ out, int n) {
  const int d = d0 + d1 + d2;
  long long idx = (long long)blockIdx.x * blockDim.x + threadIdx.x;
  if (idx >= (long long)n * d) return;
  int i = (int)(idx / d);
  int j = (int)(idx % d);
  float v;
  if (j < d0)            v = s0[(long long)i * d0 + j];
  else if (j < d0 + d1)  v = s1[(long long)i * d1 + (j - d0)];
  else                   v = s2[(long long)i * d2 + (j - d0 - d1)];
  out[idx] = (v - rm[j]) * (g[j] * rsqrtf(rv[j] + BN_EPS)) + b[j];
}

// ---------------- fp32 WMMA GEMM: C[M x NT*16] = A[M x din] * W[din x NT*16] ----------
//
// Block = 256 threads = 8 waves; wave w owns M-tile (blockIdx.x*8 + w) and accumulates
// ALL NT column tiles (register blocking: NT x v8f accumulators). W K-chunks are staged
// into LDS once per block (async load-to-LDS when available) and read back as B
// fragments via ds_load; A fragments are read from global exactly once.
//
// Fragment layouts for V_WMMA_F32_16X16X4_F32 (wave32):
//   A (16x4): lanes 0-15 row M=lane, {K,K+1}; lanes 16-31 {K+2,K+3}.
//   B (4x16): v0 = row K (lo half) / K+2 (hi half), v1 = K+1 / K+3, col = lane&15.
//   C/D (16x16): VGPR g -> M=g (lanes 0-15) / M=g+8 (lanes 16-31), N = lane&15.

template <int NT>
__global__ void __launch_bounds__(256)
wmma_gemm_f32(const float* __restrict__ A, const float* __restrict__ W,
              float* __restrict__ C, int M, int din) {
  constexpr int DOUT = NT * 16;
  constexpr int LSTR = DOUT + 8;  // padded stride: halves of the wave hit disjoint banks
  __shared__ float bpanel[KC * LSTR];

  const int lane = threadIdx.x & 31;
  const int wave = threadIdx.x >> 5;
  const int mTiles = M >> 4;
  int mTile = blockIdx.x * 8 + wave;
  if (mTile >= mTiles) mTile = mTiles - 1;  // clamp: keeps barriers + EXEC wave-uniform

  const int half = lane >> 4;  // 0: lanes 0-15, 1: lanes 16-31
  const int l16  = lane & 15;
  const float* Arow = A + (long long)(mTile * 16 + l16) * din;

  v8f acc[NT];
#pragma unroll
  for (int t = 0; t < NT; ++t) acc[t] = (v8f){};

  for (int k0 = 0; k0 < din; k0 += KC) {
    // ---- stage W[k0 .. k0+KC) x [0 .. DOUT) into LDS, float4 chunks ----
    constexpr int CHUNKS = KC * DOUT / 4;  // multiple of 256 -> uniform trip count
    for (int c = threadIdx.x; c < CHUNKS; c += 256) {
      const int krow = c / (DOUT / 4);
      const int c4   = (c % (DOUT / 4)) * 4;
      const float* gsrc = W + (long long)(k0 + krow) * DOUT + c4;
      float* ldst = &bpanel[krow * LSTR + c4];
#if USE_ASYNC_LDS
      __builtin_amdgcn_global_load_async_to_lds_b128(
          (gas_v4i_ptr)(void*)gsrc, (las_v4i_ptr)(void*)ldst, 0, 0);
#else
      *(float4*)ldst = *(const float4*)gsrc;
#endif
    }
#if USE_ASYNC_LDS
    __builtin_amdgcn_s_wait_asynccnt(0);
#endif
    __syncthreads();

    // ---- K-loop over this chunk: 1 A-frag load feeds NT WMMAs ----
    for (int k = 0; k < KC; k += 4) {
      const int kk = k + 2 * half;
      v2f a;
      a.x = Arow[k0 + kk + 0];
      a.y = Arow[k0 + kk + 1];
      const float* brow0 = &bpanel[(kk + 0) * LSTR + l16];
      const float* brow1 = &bpanel[(kk + 1) * LSTR + l16];
#pragma unroll
      for (int t = 0; t < NT; ++t) {
        v2f b;
        b.x = brow0[t * 16];
        b.y = brow1[t * 16];
        acc[t] = __builtin_amdgcn_wmma_f32_16x16x4_f32(
            /*neg_a=*/false, a, /*neg_b=*/false, b,
            /*c_mod=*/(short)0, acc[t], /*reuse_a=*/false, /*reuse_b=*/false);
      }
    }
    __syncthreads();
  }

  // ---- store all NT tiles ----
  float* Cbase = C + (long long)(mTile * 16 + half * 8) * DOUT + l16;
#pragma unroll
  for (int t = 0; t < NT; ++t)
#pragma unroll
    for (int g = 0; g < 8; ++g)
      Cbase[(long long)g * DOUT + t * 16] = acc[t][g];
}

// ---------------- aggregation: self-loop + bias init, edge scatter, relu --------------

__global__ void agg_init(const float* __restrict__ hw, const float* __restrict__ dinv,
                         const float* __restrict__ bias, float* __restrict__ agg,
                         int n, int d) {
  long long idx = (long long)blockIdx.x * blockDim.x + threadIdx.x;
  if (idx >= (long long)n * d) return;
  int i = (int)(idx / d);
  int j = (int)(idx % d);
  float s = dinv[i];
  agg[idx] = hw[idx] * s * s + bias[j];
}

__global__ void scatter_edges(const int* __restrict__ row, const int* __restrict__ col,
                              const float* __restrict__ nrm, const float* __restrict__ hw,
                              float* __restrict__ agg, int e, int d) {
  const int chunks = d >> 2;  // float4 chunks per edge
  long long idx = (long long)blockIdx.x * blockDim.x + threadIdx.x;
  if (idx >= (long long)e * chunks) return;
  int ed = (int)(idx / chunks);
  int c  = ((int)(idx % chunks)) << 2;
  int r  = row[ed];
  int cl = col[ed];
  float nm = nrm[ed];
  float4 v = *(const float4*)(hw + (long long)r * d + c);
  float* dst = agg + (long long)cl * d + c;
  atomicAdd(dst + 0, v.x * nm);
  atomicAdd(dst + 1, v.y * nm);
  atomicAdd(dst + 2, v.z * nm);
  atomicAdd(dst + 3, v.w * nm);
}

__global__ void relu_inplace(float* __restrict__ p, long long n) {
  long long i = (long long)blockIdx.x * blockDim.x + threadIdx.x;
  if (i < n) p[i] = fmaxf(p[i], 0.0f);
}

// ---------------- host launch ----------------

static inline int nblk(long long total, int bs) { return (int)((total + bs - 1) / bs); }

extern "C" void kernel_launch(void* const* d_in, const int* in_sizes, int n_in,
                              void* d_out, int out_size, void* d_ws, size_t ws_size,
                              hipStream_t stream) {
  (void)in_sizes; (void)n_in; (void)out_size; (void)ws_size;

  const float* x    = (const float*)d_in[0];
  const int*   ei   = (const int*)d_in[1];
  const int*   row  = ei;
  const int*   colp = ei + N_EDGES;
  // d_in[2] percent, d_in[3] ricci_curvature: unused in eval path
  const float* W1   = (const float*)d_in[4];
  const float* b1   = (const float*)d_in[5];
  const float* Wc0  = (const float*)d_in[6];
  const float* bc0  = (const float*)d_in[7];
  const float* Wc1  = (const float*)d_in[8];
  const float* bc1  = (const float*)d_in[9];
  const float* Wout = (const float*)d_in[10];
  const float* bout = (const float*)d_in[11];
  const float* bn1[4]  = {(const float*)d_in[12], (const float*)d_in[13],
                          (const float*)d_in[14], (const float*)d_in[15]};
  const float* bnc0[4] = {(const float*)d_in[16], (const float*)d_in[17],
                          (const float*)d_in[18], (const float*)d_in[19]};
  const float* bnc1[4] = {(const float*)d_in[20], (const float*)d_in[21],
                          (const float*)d_in[22], (const float*)d_in[23]};
  const float* bn2[4]  = {(const float*)d_in[24], (const float*)d_in[25],
                          (const float*)d_in[26], (const float*)d_in[27]};

  float* out = (float*)d_out;                       // [N, 64]
  float* r0  = out + (long long)N_NODES * 64;       // [N, 128]
  float* r1  = r0 + (long long)N_NODES * 128;       // [N, 128]
  float* r2  = r1 + (long long)N_NODES * 128;       // [N, 128]

  float* ws   = (float*)d_ws;
  float* dinv = ws;                                  // [N]
  float* nrm  = dinv + N_NODES;                      // [E]
  float* hbn  = nrm + N_EDGES;                       // [N, 384] max
  float* hw   = hbn + (long long)N_NODES * 384;      // [N, 128] max

  const int BS = 256;

  // GCN normalization (shared by all layers)
  deg_init<<<nblk(N_NODES, BS), BS, 0, stream>>>(dinv, N_NODES);
  deg_accum<<<nblk(N_EDGES, BS), BS, 0, stream>>>(colp, dinv, N_EDGES);
  deg_fin<<<nblk(N_NODES, BS), BS, 0, stream>>>(dinv, N_NODES);
  edge_norm<<<nblk(N_EDGES, BS), BS, 0, stream>>>(row, colp, dinv, nrm, N_EDGES);

  auto layer = [&](const float* s0, const float* s1, const float* s2,
                   int d0, int d1, int d2,
                   const float* W, const float* bias, const float* const* bn,
                   float* dst, int dout_, bool relu) {
    const int din = d0 + d1 + d2;
    const long long totIn  = (long long)N_NODES * din;
    const long long totOut = (long long)N_NODES * dout_;
    bn_eval_concat<<<nblk(totIn, BS), BS, 0, stream>>>(
        s0, s1, s2, d0, d1, d2, bn[0], bn[1], bn[2], bn[3], hbn, N_NODES);
    const int mTiles = N_NODES / 16;           // 3125
    const int gemmBlocks = (mTiles + 7) / 8;   // 8 M-tiles (waves) per block
    if (dout_ == 128)
      wmma_gemm_f32<8><<<gemmBlocks, BS, 0, stream>>>(hbn, W, hw, N_NODES, din);
    else
      wmma_gemm_f32<4><<<gemmBlocks, BS, 0, stream>>>(hbn, W, hw, N_NODES, din);
    agg_init<<<nblk(totOut, BS), BS, 0, stream>>>(hw, dinv, bias, dst, N_NODES, dout_);
    const long long totE = (long long)N_EDGES * (dout_ / 4);
    scatter_edges<<<nblk(totE, BS), BS, 0, stream>>>(row, colp, nrm, hw, dst, N_EDGES, dout_);
    if (relu) relu_inplace<<<nblk(totOut, BS), BS, 0, stream>>>(dst, totOut);
  };

  // layer 1: bn1(x) -> gcn(W1) -> relu -> r0
  layer(x, nullptr, nullptr, 128, 0, 0, W1, b1, bn1, r0, 128, true);
  // layer 2: bnc0(r0) -> gcn(Wc0) -> relu -> r1
  layer(r0, nullptr, nullptr, 128, 0, 0, Wc0, bc0, bnc0, r1, 128, true);
  // layer 3: bnc1([r0|r1]) -> gcn(Wc1) -> relu -> r2
  layer(r0, r1, nullptr, 128, 128, 0, Wc1, bc1, bnc1, r2, 128, true);
  // output layer: bn2([r0|r1|r2]) -> gcn(Wout) -> out (no relu)
  layer(r0, r1, r2, 128, 128, 128, Wout, bout, bn2, out, 64, false);
}